// RateBasedNeuron_3642132267483
// MI455X (gfx1250) — compile-verified
//
#include <hip/hip_runtime.h>
#include <cstdint>

typedef float v2f __attribute__((ext_vector_type(2)));
typedef float v8f __attribute__((ext_vector_type(8)));

#define HAS_WMMA4 __has_builtin(__builtin_amdgcn_wmma_f32_16x16x4_f32)
#define USE_ASYNC_LDS 1

// ---------------- helpers ----------------

// w^e via square-and-multiply (deterministic)
__device__ __forceinline__ float powi_u(float b, unsigned e) {
    float r = 1.0f;
    while (e) { if (e & 1u) r *= b; b *= b; e >>= 1u; }
    return r;
}

// w^e for e <= 15; 0 for e < 0. Fully branchless: cndmask/mul chain + final select,
// so no EXEC-mask branches appear between the WMMAs that consume these values.
__device__ __forceinline__ float powi_small(float w, int e) {
    float v = 1.0f;
#pragma unroll
    for (int j = 0; j < 15; ++j) v *= (j < e) ? w : 1.0f;
    return (e < 0) ? 0.0f : v;
}

// 3-state refractory map, encoded 2 bits/state: bits[2s+1:2s] = map(s)
#define MAP_ID 0x24u   // identity: 0,1,2
__device__ __forceinline__ unsigned map_apply(unsigned m, unsigned s) {
    return (m >> (2u * s)) & 3u;
}
__device__ __forceinline__ unsigned map_then(unsigned x, unsigned y) { // apply x, then y
    unsigned r0 = map_apply(y, map_apply(x, 0u));
    unsigned r1 = map_apply(y, map_apply(x, 1u));
    unsigned r2 = map_apply(y, map_apply(x, 2u));
    return r0 | (r1 << 2) | (r2 << 4);
}
__device__ __forceinline__ unsigned map_from_bits(unsigned bits, int n) {
    unsigned m0 = 0u, m1 = 1u, m2 = 2u;
#pragma unroll
    for (int j = 0; j < 16; ++j) {
        if (j >= n) break;
        unsigned b = (bits >> j) & 1u;
        m0 = m0 > 0u ? m0 - 1u : (b ? 2u : 0u);
        m1 = m1 > 0u ? m1 - 1u : (b ? 2u : 0u);
        m2 = m2 > 0u ? m2 - 1u : (b ? 2u : 0u);
    }
    return m0 | (m1 << 2) | (m2 << 4);
}

#if USE_ASYNC_LDS
__device__ __forceinline__ void async_g2l_b128(uint32_t lds_byte_off, const float* gaddr) {
    asm volatile("global_load_async_to_lds_b128 %0, %1, off"
                 :: "v"(lds_byte_off), "v"(gaddr) : "memory");
}
__device__ __forceinline__ void wait_async0() {
    asm volatile("s_wait_asynccnt 0" ::: "memory");
}
#endif

// ---------------- K1: per-4096-chunk affine summary B_chunk ----------------
// B_chunk = sum_{t in chunk} alpha*I_t * w^{4095 - t_local}   (b_0 := 0 globally)
__global__ void __launch_bounds__(256)
k1_chunk_sum(const float* __restrict__ I, const float* __restrict__ tauP,
             float* __restrict__ Bc) {
    __shared__ float ls[256];
    const int t = threadIdx.x;
    const float tau = tauP[0];
    const float alpha = 1.0f / tau;       // DT = 1
    const float w = 1.0f - alpha;
    const int base = blockIdx.x * 4096 + t * 16;
    const float4* I4 = (const float4*)I;
    float s = 0.0f;
#pragma unroll
    for (int q = 0; q < 4; ++q) {
        float4 v = I4[(base >> 2) + q];
        if (blockIdx.x == 0 && t == 0 && q == 0) v.x = 0.0f;  // rate[0] = 0
        s = s * w + alpha * v.x;
        s = s * w + alpha * v.y;
        s = s * w + alpha * v.z;
        s = s * w + alpha * v.w;
    }
    const float w16 = powi_u(w, 16u);
    s *= powi_u(w16, (unsigned)(255 - t));
    ls[t] = s;
    __syncthreads();
    for (int st = 128; st > 0; st >>= 1) {
        if (t < st) ls[t] += ls[t + st];
        __syncthreads();
    }
    if (t == 0) Bc[blockIdx.x] = ls[0];
}

// ---------------- K2: scan chunk affines -> r_in per chunk ----------------
__global__ void __launch_bounds__(256)
k2_scan_rate(const float* __restrict__ Bc, const float* __restrict__ tauP,
             float* __restrict__ rin, int nC) {
    __shared__ float lA[256];
    __shared__ float lB[256];
    const int t = threadIdx.x;
    const float tau = tauP[0];
    const float w = 1.0f - 1.0f / tau;
    const float a = powi_u(w, 4096u);     // underflows to 0 for tau=20, handled generically
    const int L = (nC + 255) / 256;
    float Ath = 1.0f, Bth = 0.0f;
    for (int j = 0; j < L; ++j) {
        int idx = t * L + j;
        if (idx < nC) { Ath *= a; Bth = Bth * a + Bc[idx]; }
    }
    lA[t] = Ath; lB[t] = Bth;
    __syncthreads();
    for (int s = 1; s < 256; s <<= 1) {
        float ca = lA[t], cb = lB[t];
        float pa = 1.0f, pb = 0.0f;
        if (t >= s) { pa = lA[t - s]; pb = lB[t - s]; }
        __syncthreads();
        lA[t] = pa * ca;
        lB[t] = pb * ca + cb;
        __syncthreads();
    }
    float r = (t == 0) ? 0.0f : lB[t - 1];   // exclusive prefix applied to r_init = 0
    for (int j = 0; j < L; ++j) {
        int idx = t * L + j;
        if (idx < nC) { rin[idx] = r; r = a * r + Bc[idx]; }
    }
}

// ---------------- K3: rates (WMMA), candidate bits, per-chunk state map ----
// Block = 512 threads (16 waves), chunk = 4096 elems; wave tile = 256 elems
// (16 segments x 16 steps). r = (alpha*L) x I_tile + C, C = w^{i+1} * r_in_seg.
__global__ void __launch_bounds__(512)
k3_rates(const float* __restrict__ I, const float* __restrict__ u,
         const float* __restrict__ tauP, const float* __restrict__ thrP,
         const float* __restrict__ rinChunk,
         unsigned* __restrict__ blockMaps, unsigned char* __restrict__ bits) {
    __shared__ float ldsI[4096];
    __shared__ float ldsT[512];     // alpha-scaled half-segment sums (w^8 folded for even lanes)
    __shared__ float ldsW[512];     // weighted contributions to tile summary
    __shared__ float ldsBtile[16];
    __shared__ float ldsRinTile[16];
    __shared__ float ldsRinSeg[256];
    __shared__ unsigned ldsMap[512];

    const int tid  = threadIdx.x;
    const int lane = tid & 31;
    const int wave = tid >> 5;
    const int cbase = blockIdx.x * 4096;
    const float tau = tauP[0], thr = thrP[0];
    const float alpha = 1.0f / tau;
    const float w   = 1.0f - alpha;
    const float w8  = powi_u(w, 8u);
    const float w16 = w8 * w8;
    const float w256 = powi_u(w16, 16u);
    const float invthr = 1.0f / thr;

    const float* Ichunk = I + cbase;
    const int eo0 = wave * 256 + lane * 8;     // this lane's 8 contiguous elements

    // ---- stage I tile into LDS (CDNA5 async DMA path) ----
#if USE_ASYNC_LDS
    {
        uint32_t lbase = (uint32_t)(size_t)(void*)&ldsI[0];
        async_g2l_b128(lbase + (uint32_t)((eo0 + 0) * 4), Ichunk + eo0);
        async_g2l_b128(lbase + (uint32_t)((eo0 + 4) * 4), Ichunk + eo0 + 4);
        wait_async0();
    }
#else
    {
        const float4* g4 = (const float4*)(Ichunk + eo0);
        *(float4*)&ldsI[eo0]     = g4[0];
        *(float4*)&ldsI[eo0 + 4] = g4[1];
    }
#endif
    if (blockIdx.x == 0 && tid == 0) ldsI[0] = 0.0f;   // b_0 = 0 (rate[0]=0); same-wave readers

    // ---- half-segment Horner sums (raw I; alpha folded below) ----
    float s = 0.0f;
#pragma unroll
    for (int j = 0; j < 8; ++j) s = s * w + ldsI[eo0 + j];
    const float tpart = alpha * s * ((lane & 1) ? 1.0f : w8);
    ldsT[tid] = tpart;
    ldsW[tid] = tpart * powi_u(w16, (unsigned)(15 - (lane >> 1)));
    __syncthreads();

    // ---- tile summaries and carry through 16 wave-tiles ----
    if (tid < 16) {
        float sum = 0.0f;
        for (int l = 0; l < 32; ++l) sum += ldsW[tid * 32 + l];
        ldsBtile[tid] = sum;
    }
    __syncthreads();
    if (tid == 0) {
        float r = rinChunk[blockIdx.x];
        for (int wv = 0; wv < 16; ++wv) { ldsRinTile[wv] = r; r = w256 * r + ldsBtile[wv]; }
    }
    __syncthreads();
    if (lane == 0) {
        float r = ldsRinTile[wave];
        for (int c = 0; c < 16; ++c) {
            ldsRinSeg[wave * 16 + c] = r;
            r = w16 * r + (ldsT[wave * 32 + 2 * c] + ldsT[wave * 32 + 2 * c + 1]);
        }
    }
    __syncthreads();

    // ---- per-lane tile coordinates ----
    const int col = lane & 15;        // segment / D column
    const int hi  = lane >> 4;        // 0: rows 0..7 (K 0,1); 1: rows 8..15 (K 2,3)
    const float rinseg = ldsRinSeg[wave * 16 + col];

    float rv[8];
#if HAS_WMMA4
    {
        // C accumulator: C[v] = rinseg * w^{v+1} (hi=0) or w^{v+9} (hi=1)
        v8f acc;
        {
            float c2 = rinseg * (hi ? (w8 * w) : w);
#pragma unroll
            for (int v = 0; v < 8; ++v) { acc[v] = c2; c2 *= w; }
        }
        const int rrow  = lane & 15;
        const int kbase = hi ? 2 : 0;
#pragma unroll
        for (int k = 0; k < 4; ++k) {
            v2f a, b;
            int e0 = rrow - (4 * k + kbase);
            a.x = alpha * powi_small(w, e0);
            a.y = alpha * powi_small(w, e0 - 1);
            int boff = wave * 256 + col * 16 + 4 * k + kbase;
            b.x = ldsI[boff];
            b.y = ldsI[boff + 1];
            acc = __builtin_amdgcn_wmma_f32_16x16x4_f32(
                false, a, false, b, (short)0, acc, false, false);
        }
#pragma unroll
        for (int v = 0; v < 8; ++v) rv[v] = acc[v];
    }
#else
    {
        // VALU fallback: walk the segment recurrence, keep the relevant half
        float r = rinseg;
#pragma unroll
        for (int j = 0; j < 16; ++j) {
            r = w * r + alpha * ldsI[wave * 256 + col * 16 + j];
            int rel = j - hi * 8;
            if (rel >= 0 && rel < 8) rv[rel] = r;
        }
    }
#endif

    // ---- fire candidates at elements eb..eb+7 ----
    const int eb = wave * 256 + col * 16 + hi * 8;
    const float4* uu4 = (const float4*)(u + cbase + eb);
    float4 ua = uu4[0], ub = uu4[1];
    float uvals[8] = {ua.x, ua.y, ua.z, ua.w, ub.x, ub.y, ub.z, ub.w};
    unsigned bitsv = 0u;
#pragma unroll
    for (int v = 0; v < 8; ++v) {
        float p = fminf(fmaxf(rv[v] * invthr, 0.0f), 1.0f) * 0.1f;  // DT/10
        bitsv |= (uvals[v] < p) ? (1u << v) : 0u;
    }
    bits[(unsigned)(cbase + eb) >> 3] = (unsigned char)bitsv;

    // ---- per-lane 3-state map, time-ordered compose across block ----
    ldsMap[wave * 32 + col * 2 + hi] = map_from_bits(bitsv, 8);
    __syncthreads();
    for (int st = 1; st < 512; st <<= 1) {
        unsigned combined = 0u;
        bool act = ((tid & (2 * st - 1)) == 0);
        if (act) combined = map_then(ldsMap[tid], ldsMap[tid + st]);
        __syncthreads();
        if (act) ldsMap[tid] = combined;
        __syncthreads();
    }
    if (tid == 0) blockMaps[blockIdx.x] = ldsMap[0];
}

// ---------------- K4: scan chunk maps -> refrac state_in per chunk ----------
__global__ void __launch_bounds__(256)
k4_scan_map(const unsigned* __restrict__ maps, unsigned* __restrict__ stin, int nC) {
    __shared__ unsigned lm[256];
    const int t = threadIdx.x;
    const int L = (nC + 255) / 256;
    unsigned M = MAP_ID;
    for (int j = 0; j < L; ++j) {
        int idx = t * L + j;
        if (idx < nC) M = map_then(M, maps[idx]);
    }
    lm[t] = M;
    __syncthreads();
    for (int s = 1; s < 256; s <<= 1) {
        unsigned cur = lm[t];
        unsigned prev = (t >= s) ? lm[t - s] : MAP_ID;
        __syncthreads();
        lm[t] = (t >= s) ? map_then(prev, cur) : cur;
        __syncthreads();
    }
    unsigned E = (t == 0) ? MAP_ID : lm[t - 1];
    unsigned st_ = map_apply(E, 0u);               // global initial refrac = 0
    for (int j = 0; j < L; ++j) {
        int idx = t * L + j;
        if (idx < nC) { stin[idx] = st_; st_ = map_apply(maps[idx], st_); }
    }
}

// ---------------- K5: emit spikes from candidate bits ----------------------
__global__ void __launch_bounds__(256)
k5_emit(const unsigned char* __restrict__ bits, const unsigned* __restrict__ stin,
        float* __restrict__ out) {
    __shared__ unsigned lm[256];
    const int t = threadIdx.x;
    const int cbase = blockIdx.x * 4096;
    const unsigned short* b16 = (const unsigned short*)(bits + (cbase >> 3));
    const unsigned bb = (unsigned)b16[t];
    lm[t] = map_from_bits(bb, 16);
    __syncthreads();
    for (int s = 1; s < 256; s <<= 1) {
        unsigned cur = lm[t];
        unsigned prev = (t >= s) ? lm[t - s] : MAP_ID;
        __syncthreads();
        lm[t] = (t >= s) ? map_then(prev, cur) : cur;
        __syncthreads();
    }
    unsigned E = (t == 0) ? MAP_ID : lm[t - 1];
    unsigned st_ = map_apply(E, stin[blockIdx.x]);
    float buf[16];
#pragma unroll
    for (int j = 0; j < 16; ++j) {
        unsigned bit = (bb >> j) & 1u;
        unsigned fire = (st_ == 0u) ? bit : 0u;
        buf[j] = fire ? 1.0f : 0.0f;
        st_ = st_ > 0u ? st_ - 1u : (fire ? 2u : 0u);
    }
    float4* o4 = (float4*)(out + cbase + t * 16);
#pragma unroll
    for (int q = 0; q < 4; ++q)
        o4[q] = make_float4(buf[4 * q], buf[4 * q + 1], buf[4 * q + 2], buf[4 * q + 3]);
}

// ---------------- host entry ----------------------------------------------
extern "C" void kernel_launch(void* const* d_in, const int* in_sizes, int n_in,
                              void* d_out, int out_size, void* d_ws, size_t ws_size,
                              hipStream_t stream) {
    (void)n_in; (void)out_size; (void)ws_size;
    const float* I   = (const float*)d_in[0];
    const float* u   = (const float*)d_in[1];
    const float* tau = (const float*)d_in[2];
    const float* thr = (const float*)d_in[3];
    float* out = (float*)d_out;

    const int N  = in_sizes[0];
    const int nC = N >> 12;           // N / 4096 (harness N = 2^23 -> 2048 chunks)
    if (nC <= 0 || (nC << 12) != N) return;

    float*    Bc   = (float*)d_ws;                      // nC floats
    float*    rin  = Bc + nC;                           // nC floats
    unsigned* maps = (unsigned*)(rin + nC);             // nC u32
    unsigned* stin = maps + nC;                         // nC u32
    unsigned char* bits = (unsigned char*)(stin + nC);  // N/8 bytes

    k1_chunk_sum<<<nC, 256, 0, stream>>>(I, tau, Bc);
    k2_scan_rate<<<1, 256, 0, stream>>>(Bc, tau, rin, nC);
    k3_rates<<<nC, 512, 0, stream>>>(I, u, tau, thr, rin, maps, bits);
    k4_scan_map<<<1, 256, 0, stream>>>(maps, stin, nC);
    k5_emit<<<nC, 256, 0, stream>>>(bits, stin, out);
}